// MultiScaleRetention_27109833572753
// MI455X (gfx1250) — compile-verified
//
#include <hip/hip_runtime.h>
#include <hip/hip_bf16.h>

// MultiScaleRetention on MI455X (gfx1250), fp32 end-to-end via V_WMMA_F32_16X16X4_F32.
//
// Pipeline:
//   1) qkv = x @ w_qkv                      (WMMA f32 GEMM, 4096x6144x2048)
//   2) xpos rotary applied in-place to q,k  (elementwise)
//   3) retention per (b,h,q-tile)           (WMMA q.k^T, decay mask, WMMA S.V)
//   4) g = x @ w_g                          (WMMA f32 GEMM, 4096x2048x2048)
//   5) z = silu(g) * groupnorm(y)           (shfl reductions, fused)
//   6) out = z @ w_proj                     (WMMA f32 GEMM, 4096x2048x2048)
//
// GEMM is double-buffered: LDS ping-pong + register staging of the next K
// chunk + global_prefetch of the chunk after that.

typedef __attribute__((ext_vector_type(2))) float v2f;
typedef __attribute__((ext_vector_type(4))) float v4f;
typedef __attribute__((ext_vector_type(8))) float v8f;

#define NHEADS 16
#define SEQ_T 1024
#define DIM_C 2048
#define HEAD_D 128
#define C3 (3 * DIM_C)

// ---------------------------------------------------------------------------
// Generic fp32 WMMA GEMM: C[M,N] = A[M,K] * B[K,N], row-major, dims % tile == 0
// Block tile 128x128, K chunk 32. 8 waves, each wave computes 64x32 (4x2 WMMA
// tiles). wave32: lane%16 = matrix row/col index, lane/16 = K-pair half.
// ---------------------------------------------------------------------------
#define LDA_S 40   // 128x32 A tile, row stride 40 floats (160B, 16B aligned)
#define LDB_S 136  // 32x128 B tile, row stride 136 floats (544B, 16B aligned)

__global__ __launch_bounds__(256) void gemm_f32_wmma_kernel(
    const float* __restrict__ A, const float* __restrict__ B,
    float* __restrict__ C, int M, int N, int K) {
  __shared__ float As[2][128 * LDA_S];
  __shared__ float Bs[2][32 * LDB_S];

  const int tid = threadIdx.x;
  const int lane = tid & 31;
  const int wave = tid >> 5;
  const int l16 = lane & 15;
  const int lhalf = lane >> 4;

  const int block_m = blockIdx.y * 128;
  const int block_n = blockIdx.x * 128;

  const int wm = (wave >> 2) * 64;  // 0 or 64
  const int wn = (wave & 3) * 32;   // 0,32,64,96

  // Per-thread staging coordinates (4 x float4 each for A and B tiles)
  int a_row[4], a_col[4], b_row[4], b_col[4];
#pragma unroll
  for (int i = 0; i < 4; ++i) {
    int idx = tid + i * 256;
    a_row[i] = idx >> 3;
    a_col[i] = (idx & 7) << 2;
    b_row[i] = idx >> 5;
    b_col[i] = (idx & 31) << 2;
  }

  v8f acc[4][2];
#pragma unroll
  for (int i = 0; i < 4; ++i)
#pragma unroll
    for (int j = 0; j < 2; ++j)
#pragma unroll
      for (int r = 0; r < 8; ++r) acc[i][j][r] = 0.0f;

  // Preload chunk 0 into registers
  v4f areg[4], breg[4];
#pragma unroll
  for (int i = 0; i < 4; ++i) {
    areg[i] = *(const v4f*)(A + (size_t)(block_m + a_row[i]) * K + a_col[i]);
    breg[i] = *(const v4f*)(B + (size_t)b_row[i] * N + (block_n + b_col[i]));
  }

  int p = 0;
  for (int k0 = 0; k0 < K; k0 += 32, p ^= 1) {
    // Commit staged registers to LDS buffer p
#pragma unroll
    for (int i = 0; i < 4; ++i) {
      *(v4f*)&As[p][a_row[i] * LDA_S + a_col[i]] = areg[i];
      *(v4f*)&Bs[p][b_row[i] * LDB_S + b_col[i]] = breg[i];
    }
    __syncthreads();

    // Stage chunk k0+32 into registers while buffer p is being consumed
    const int kn = k0 + 32;
    if (kn < K) {
#pragma unroll
      for (int i = 0; i < 4; ++i) {
        areg[i] =
            *(const v4f*)(A + (size_t)(block_m + a_row[i]) * K + kn + a_col[i]);
        breg[i] =
            *(const v4f*)(B + (size_t)(kn + b_row[i]) * N + block_n + b_col[i]);
      }
    }
    // Pull chunk k0+64 toward cache (global_prefetch_b8)
    if (k0 + 64 < K) {
      __builtin_prefetch(A + (size_t)(block_m + a_row[0]) * K + k0 + 64 + a_col[0],
                         0, 3);
      __builtin_prefetch(
          B + (size_t)(k0 + 64 + b_row[0]) * N + block_n + b_col[0], 0, 3);
    }

    // Consume buffer p: 8 K-steps x (4x2) WMMA tiles
#pragma unroll
    for (int kk = 0; kk < 32; kk += 4) {
      // A fragments: lane holds A[m=l16][k = kk + 2*lhalf + {0,1}]
      v2f af[4];
#pragma unroll
      for (int i = 0; i < 4; ++i)
        af[i] =
            *(const v2f*)&As[p][(wm + i * 16 + l16) * LDA_S + kk + 2 * lhalf];
      // B fragments: lane holds B[k = kk + 2*lhalf + {0,1}][n=l16]
      v2f bf[2];
#pragma unroll
      for (int j = 0; j < 2; ++j) {
        int col = wn + j * 16 + l16;
        int row = kk + 2 * lhalf;
        v2f t;
        t.x = Bs[p][row * LDB_S + col];
        t.y = Bs[p][(row + 1) * LDB_S + col];
        bf[j] = t;
      }
#pragma unroll
      for (int i = 0; i < 4; ++i)
#pragma unroll
        for (int j = 0; j < 2; ++j)
          acc[i][j] = __builtin_amdgcn_wmma_f32_16x16x4_f32(
              false, af[i], false, bf[j], (short)0, acc[i][j], false, false);
    }
    // No trailing barrier needed: next iteration writes the other buffer, and
    // its barrier (after the writes) fences reuse of this one.
  }

  // Store: D layout -> lane holds D[m = r + 8*lhalf][n = l16] in VGPR r
#pragma unroll
  for (int i = 0; i < 4; ++i) {
    int row_base = block_m + wm + i * 16 + lhalf * 8;
#pragma unroll
    for (int j = 0; j < 2; ++j) {
      int col = block_n + wn + j * 16 + l16;
#pragma unroll
      for (int r = 0; r < 8; ++r)
        C[(size_t)(row_base + r) * N + col] = acc[i][j][r];
    }
  }
}

// ---------------------------------------------------------------------------
// xPos rotary, applied in-place to the q and k slices of the qkv buffer.
// One thread per (element pair); first half of grid covers q, second covers k.
// ---------------------------------------------------------------------------
__global__ __launch_bounds__(256) void xpos_kernel(float* __restrict__ qkv,
                                                   int BT /* = B*T */) {
  const int halfC = DIM_C / 2;
  size_t i = (size_t)blockIdx.x * blockDim.x + threadIdx.x;
  size_t per_tensor = (size_t)BT * halfC;
  if (i >= per_tensor * 2) return;
  int which = (i >= per_tensor) ? 1 : 0;  // 0=q, 1=k
  size_t p = which ? (i - per_tensor) : i;
  int c2 = (int)(p % halfC);
  size_t bt = p / halfC;
  int t = (int)(bt % SEQ_T);

  float pos = (float)t;
  float ds = ((float)(2 * c2) + 0.4f * DIM_C) / (1.4f * DIM_C);
  float sc = exp2f(log2f(ds) * (pos * (1.0f / 512.0f)));
  // 1/10000^(2c2/C) ; log2(10000) = 13.287712379549449
  float freq = exp2f(-((float)(2 * c2) / (float)DIM_C) * 13.287712379549449f);
  float ang = pos * freq;
  float sn = sinf(ang) * sc;
  float cs = cosf(ang) * sc;

  float* base = qkv + bt * C3 + (size_t)which * DIM_C + 2 * c2;
  float x0 = base[0], x1 = base[1];
  base[0] = x0 * cs - x1 * sn;  // even slot: x0*cos + (-x1)*sin
  base[1] = x1 * cs + x0 * sn;  // odd  slot: x1*cos + ( x0)*sin
}

// ---------------------------------------------------------------------------
// Retention: one wave (32 threads) per (b, h, 16-row q tile).
// Q kept in registers as 32 A-fragments. Causal loop over key tiles:
//   S = Q.K^T (32 wmma) -> decay mask -> LDS transpose -> Y += S.V (32 wmma)
// y stored as (B, H, T, D). Single-wave blocks: __syncthreads() is only a
// wave-local ordering fence for the LDS transpose (barriers lower to s_nop
// for 1-wave workgroups, but keep the compiler from reordering LDS ops).
// ---------------------------------------------------------------------------
__global__ __launch_bounds__(32) void retention_kernel(
    const float* __restrict__ qkv, float* __restrict__ y) {
  const int bid = blockIdx.x;
  const int qt = bid & 63;          // T/16 = 64 q tiles
  const int h = (bid >> 6) & 15;
  const int b = bid >> 10;
  const int lane = threadIdx.x & 31;
  const int l16 = lane & 15;
  const int lhalf = lane >> 4;

  __shared__ float Sld[16 * 20];  // S tile staging, row stride 20 (8B aligned)

  const float gamma = 1.0f - exp2f(-5.0f - (float)h);
  const float log2g = log2f(gamma);
  const float scale = 0.088388347648318447f;  // 128^-0.5

  const int q0 = qt * 16;
  const size_t row0 = (size_t)b * SEQ_T;

  const float* Qb = qkv + row0 * C3 + (size_t)h * HEAD_D;           // q slice
  const float* Kb = qkv + row0 * C3 + DIM_C + (size_t)h * HEAD_D;   // k slice
  const float* Vb = qkv + row0 * C3 + 2 * DIM_C + (size_t)h * HEAD_D;

  // Q fragments: lane holds Q[q0+l16][k = 4s + 2*lhalf + {0,1}]
  v2f qf[32];
#pragma unroll
  for (int s = 0; s < 32; ++s)
    qf[s] = *(const v2f*)(Qb + (size_t)(q0 + l16) * C3 + s * 4 + 2 * lhalf);

  v8f acc[8];
#pragma unroll
  for (int j = 0; j < 8; ++j)
#pragma unroll
    for (int r = 0; r < 8; ++r) acc[j][r] = 0.0f;

  for (int st = 0; st <= qt; ++st) {
    const int s0 = st * 16;
    // S = Q . K^T : B-fragment is K^T, lane holds K[s0+l16][k..k+1]
    v8f sacc;
#pragma unroll
    for (int r = 0; r < 8; ++r) sacc[r] = 0.0f;
#pragma unroll
    for (int s = 0; s < 32; ++s) {
      v2f bf = *(const v2f*)(Kb + (size_t)(s0 + l16) * C3 + s * 4 + 2 * lhalf);
      sacc = __builtin_amdgcn_wmma_f32_16x16x4_f32(false, qf[s], false, bf,
                                                   (short)0, sacc, false, false);
    }
    // Decay mask * D^-0.5 ; lane element (m = r + 8*lhalf, n = l16)
#pragma unroll
    for (int r = 0; r < 8; ++r) {
      int diff = (q0 + r + 8 * lhalf) - (s0 + l16);
      float mv = (diff >= 0) ? exp2f((float)diff * log2g) * scale : 0.0f;
      sacc[r] *= mv;
    }
    // C-layout -> LDS -> A-layout (transpose staging, single wave)
#pragma unroll
    for (int r = 0; r < 8; ++r) Sld[(r + 8 * lhalf) * 20 + l16] = sacc[r];
    __syncthreads();
    v2f sa[4];
#pragma unroll
    for (int s = 0; s < 4; ++s)
      sa[s] = *(const v2f*)&Sld[l16 * 20 + s * 4 + 2 * lhalf];
    __syncthreads();
    // Y += S . V : 8 n-tiles of 16, K dim = 16 (4 wmma steps each)
#pragma unroll
    for (int j = 0; j < 8; ++j) {
#pragma unroll
      for (int s = 0; s < 4; ++s) {
        int k = s * 4 + 2 * lhalf;
        v2f bf;
        bf.x = Vb[(size_t)(s0 + k) * C3 + j * 16 + l16];
        bf.y = Vb[(size_t)(s0 + k + 1) * C3 + j * 16 + l16];
        acc[j] = __builtin_amdgcn_wmma_f32_16x16x4_f32(
            false, sa[s], false, bf, (short)0, acc[j], false, false);
      }
    }
  }

  // Store Y[b][h][q0+m][d]
  float* Yb = y + ((size_t)(b * NHEADS + h) * SEQ_T) * HEAD_D;
#pragma unroll
  for (int j = 0; j < 8; ++j)
#pragma unroll
    for (int r = 0; r < 8; ++r)
      Yb[(size_t)(q0 + r + 8 * lhalf) * HEAD_D + j * 16 + l16] = acc[j][r];
}

// ---------------------------------------------------------------------------
// GroupNorm(y) over D per (b,t,h) + silu(g) gate, fused; writes z over g.
// Block = one (b,t) row, 256 threads; 16 threads per head (shfl width 16).
// ---------------------------------------------------------------------------
__global__ __launch_bounds__(256) void gn_gate_kernel(
    const float* __restrict__ y, const float* __restrict__ g,
    const float* __restrict__ gnw, const float* __restrict__ gnb,
    float* __restrict__ z) {
  const int bt = blockIdx.x;
  const int b = bt / SEQ_T;
  const int t = bt % SEQ_T;
  const int tid = threadIdx.x;
  const int h = tid >> 4;
  const int sub = tid & 15;

  const float* yh = y + ((size_t)(b * NHEADS + h) * SEQ_T + t) * HEAD_D;
  float vals[8];
  float s = 0.0f, s2 = 0.0f;
#pragma unroll
  for (int e = 0; e < 8; ++e) {
    float v = yh[sub * 8 + e];
    vals[e] = v;
    s += v;
    s2 += v * v;
  }
#pragma unroll
  for (int off = 8; off >= 1; off >>= 1) {
    s += __shfl_xor(s, off, 16);
    s2 += __shfl_xor(s2, off, 16);
  }
  const float mean = s * (1.0f / HEAD_D);
  const float var = s2 * (1.0f / HEAD_D) - mean * mean;
  const float inv = rsqrtf(var + 1e-5f);

  const size_t row = (size_t)bt * DIM_C;
#pragma unroll
  for (int e = 0; e < 8; ++e) {
    int c = h * HEAD_D + sub * 8 + e;
    float yn = (vals[e] - mean) * inv;
    yn = yn * gnw[c] + gnb[c];
    float gv = g[row + c];
    float sig = 1.0f / (1.0f + expf(-gv));
    z[row + c] = gv * sig * yn;  // silu(g) * yn
  }
}

// ---------------------------------------------------------------------------
extern "C" void kernel_launch(void* const* d_in, const int* in_sizes, int n_in,
                              void* d_out, int out_size, void* d_ws,
                              size_t ws_size, hipStream_t stream) {
  const float* x = (const float*)d_in[0];       // (B,T,C)
  const float* w_qkv = (const float*)d_in[1];   // (C, 3C)
  const float* w_g = (const float*)d_in[2];     // (C, C)
  const float* w_proj = (const float*)d_in[3];  // (C, C)
  const float* gnw = (const float*)d_in[4];     // (C,)
  const float* gnb = (const float*)d_in[5];     // (C,)
  float* out = (float*)d_out;

  const int B = 4;
  const size_t BT = (size_t)B * SEQ_T;  // 4096

  // Workspace: qkv (BT x 3C) | y (B,H,T,D) = BT*C | g/z (BT x C)
  float* qkv = (float*)d_ws;
  float* yb = qkv + BT * C3;
  float* gb = yb + BT * DIM_C;

  // 1) qkv GEMM: 4096 x 6144 x 2048
  gemm_f32_wmma_kernel<<<dim3(C3 / 128, (int)(BT / 128)), 256, 0, stream>>>(
      x, w_qkv, qkv, (int)BT, C3, DIM_C);

  // 2) xPos on q and k slices (in place)
  {
    size_t total = BT * (DIM_C / 2) * 2;
    xpos_kernel<<<(int)((total + 255) / 256), 256, 0, stream>>>(qkv, (int)BT);
  }

  // 3) retention: one wave per (b, h, q-tile)
  retention_kernel<<<B * NHEADS * (SEQ_T / 16), 32, 0, stream>>>(qkv, yb);

  // 4) gate GEMM: g = x @ w_g (4096 x 2048 x 2048)
  gemm_f32_wmma_kernel<<<dim3(DIM_C / 128, (int)(BT / 128)), 256, 0, stream>>>(
      x, w_g, gb, (int)BT, DIM_C, DIM_C);

  // 5) groupnorm + silu gate (z written over g buffer)
  gn_gate_kernel<<<(int)BT, 256, 0, stream>>>(yb, gb, gnw, gnb, gb);

  // 6) out = z @ w_proj (4096 x 2048 x 2048)
  gemm_f32_wmma_kernel<<<dim3(DIM_C / 128, (int)(BT / 128)), 256, 0, stream>>>(
      gb, w_proj, out, (int)BT, DIM_C, DIM_C);
}